// GNN_24146306138816
// MI455X (gfx1250) — compile-verified
//
#include <hip/hip_runtime.h>
#include <math.h>

// GCN, 2 layers. N=100000 nodes, F_IN=128, H=16, E edges + N self loops.
// Pipeline: init -> degree -> rsqrt -> WMMA GEMM (x@W1) -> edge scatter (16 lanes/edge)
//        -> bias+selfloop+relu -> dot16 (h@W2) -> scalar edge scatter -> sigmoid.

#define F_IN 128
#define HID  16

typedef __attribute__((ext_vector_type(2))) float v2f;
typedef __attribute__((ext_vector_type(8))) float v8f;

// ---- init workspace: deg=1 (self loop), agg1=0, agg2=0 -------------------
__global__ void k_init(float* __restrict__ deg, float* __restrict__ agg1,
                       float* __restrict__ agg2, int N) {
    int j = blockIdx.x * blockDim.x + threadIdx.x;
    if (j < N * HID) agg1[j] = 0.0f;
    if (j < N) { deg[j] = 1.0f; agg2[j] = 0.0f; }
}

// ---- degree: deg[col[e]] += 1 --------------------------------------------
__global__ void k_degree(const int* __restrict__ col, int E, float* __restrict__ deg) {
    int e = blockIdx.x * blockDim.x + threadIdx.x;
    if (e < E) atomicAdd(&deg[col[e]], 1.0f);
}

// ---- dis = rsqrt(deg), in place (deg >= 1 always) ------------------------
__global__ void k_rsqrt(float* __restrict__ deg, int N) {
    int i = blockIdx.x * blockDim.x + threadIdx.x;
    if (i < N) deg[i] = rsqrtf(deg[i]);
}

// ---- h1 = x @ W1 via V_WMMA_F32_16X16X4_F32 -------------------------------
// Block: 128 threads = 4 waves; each wave owns a 16-row x 16-col tile.
// x tile (64x128) staged in LDS with +4 padding (conflict-free A reads),
// W1 (128x16) staged in LDS.
__global__ __launch_bounds__(128)
void k_gemm1(const float* __restrict__ x, const float* __restrict__ W1,
             float* __restrict__ h1, int N) {
    __shared__ float sx[64][F_IN + 4];   // 64*132*4 = 33792 B
    __shared__ float sw[F_IN][HID];      // 8192 B

    const int t = threadIdx.x;
    const int rowBase = blockIdx.x * 64;

    // stage W1 (2048 floats)
    for (int j = t; j < F_IN * HID; j += 128)
        sw[j >> 4][j & 15] = W1[j];

    // stage x tile with coalesced float4 loads (64 rows x 32 float4)
    const float4* x4 = (const float4*)x;
    for (int j = t; j < 64 * 32; j += 128) {
        int r = j >> 5, c = j & 31;
        int gr = rowBase + r;
        float4 v = (gr < N) ? x4[(size_t)gr * 32 + c] : make_float4(0.f, 0.f, 0.f, 0.f);
        *(float4*)&sx[r][c * 4] = v;
    }
    __syncthreads();

    const int wave = t >> 5;
    const int lane = t & 31;
    const int tileRow = rowBase + wave * 16;
    if (tileRow >= N) return;            // N % 16 == 0 -> whole-tile granularity

    const int m  = lane & 15;            // A: row M, B: col N
    const int kk = (lane >> 4) << 1;     // lanes 0-15 -> K 0/1, lanes 16-31 -> K 2/3

    v8f acc = {};
    #pragma unroll
    for (int k0 = 0; k0 < F_IN; k0 += 4) {
        v2f a, b;
        a.x = sx[wave * 16 + m][k0 + kk];
        a.y = sx[wave * 16 + m][k0 + kk + 1];
        b.x = sw[k0 + kk][m];
        b.y = sw[k0 + kk + 1][m];
        // (neg_a, A, neg_b, B, c_mod, C, reuse_a, reuse_b)
        acc = __builtin_amdgcn_wmma_f32_16x16x4_f32(
                  false, a, false, b, (short)0, acc, false, false);
    }

    // C/D layout: VGPR r -> row r (lanes 0-15) / row 8+r (lanes 16-31), col = lane&15
    const int rOut = tileRow + ((lane >> 4) << 3);
    #pragma unroll
    for (int vr = 0; vr < 8; ++vr)
        h1[(size_t)(rOut + vr) * HID + m] = acc[vr];
}

// ---- layer-1 scatter: 16 lanes per edge ----------------------------------
// agg1[col][f] += h1[row][f] * dis[row]*dis[col]
__global__ void k_scatter1(const int* __restrict__ row, const int* __restrict__ col,
                           const float* __restrict__ dis, const float* __restrict__ h1,
                           float* __restrict__ agg1, int E) {
    int e = blockIdx.x * (blockDim.x >> 4) + (threadIdx.x >> 4);
    int f = threadIdx.x & 15;
    if (e >= E) return;
    int r = row[e], c = col[e];
    float nrm = dis[r] * dis[c];
    atomicAdd(&agg1[(size_t)c * HID + f], h1[(size_t)r * HID + f] * nrm);
}

// ---- self-loop + bias + relu, in place over agg1 -------------------------
__global__ void k_relu_bias(float* __restrict__ agg1, const float* __restrict__ h1,
                            const float* __restrict__ dis, const float* __restrict__ b1,
                            int N) {
    int j = blockIdx.x * blockDim.x + threadIdx.x;
    if (j >= N * HID) return;
    int i = j >> 4, f = j & 15;
    float d = dis[i];
    float v = agg1[j] + h1[j] * d * d + b1[f];
    agg1[j] = v > 0.0f ? v : 0.0f;
}

// ---- h2 = hrelu @ W2 (dot-16 per node; no bias here, bias after aggregation)
__global__ void k_gemm2(const float* __restrict__ hrelu, const float* __restrict__ W2,
                        float* __restrict__ h2, int N) {
    int i = blockIdx.x * blockDim.x + threadIdx.x;
    if (i >= N) return;
    const float4* hv = (const float4*)(hrelu + (size_t)i * HID);
    float acc = 0.0f;
    #pragma unroll
    for (int q = 0; q < 4; ++q) {
        float4 h = hv[q];
        acc += h.x * W2[q * 4 + 0] + h.y * W2[q * 4 + 1]
             + h.z * W2[q * 4 + 2] + h.w * W2[q * 4 + 3];
    }
    h2[i] = acc;
}

// ---- layer-2 scalar scatter ----------------------------------------------
__global__ void k_scatter2(const int* __restrict__ row, const int* __restrict__ col,
                           const float* __restrict__ dis, const float* __restrict__ h2,
                           float* __restrict__ agg2, int E) {
    int e = blockIdx.x * blockDim.x + threadIdx.x;
    if (e >= E) return;
    int r = row[e], c = col[e];
    atomicAdd(&agg2[c], h2[r] * dis[r] * dis[c]);
}

// ---- self-loop + bias + sigmoid ------------------------------------------
__global__ void k_final(const float* __restrict__ agg2, const float* __restrict__ h2,
                        const float* __restrict__ dis, const float* __restrict__ b2,
                        float* __restrict__ out, int N) {
    int i = blockIdx.x * blockDim.x + threadIdx.x;
    if (i >= N) return;
    float d = dis[i];
    float v = agg2[i] + h2[i] * d * d + b2[0];
    out[i] = 1.0f / (1.0f + expf(-v));
}

extern "C" void kernel_launch(void* const* d_in, const int* in_sizes, int n_in,
                              void* d_out, int out_size, void* d_ws, size_t ws_size,
                              hipStream_t stream) {
    const float* x  = (const float*)d_in[0];
    const int*   ei = (const int*)d_in[1];
    const float* W1 = (const float*)d_in[2];
    const float* b1 = (const float*)d_in[3];
    const float* W2 = (const float*)d_in[4];
    const float* b2 = (const float*)d_in[5];

    const int N = in_sizes[0] / F_IN;
    const int E = in_sizes[1] / 2;
    const int* row = ei;          // edge_index[0] = source
    const int* col = ei + E;      // edge_index[1] = target

    // workspace layout (floats): deg/dis[N] | h1[16N] | agg1[16N] | h2[N] | agg2[N]
    float* deg  = (float*)d_ws;
    float* h1   = deg  + N;
    float* agg1 = h1   + (size_t)N * HID;
    float* h2   = agg1 + (size_t)N * HID;
    float* agg2 = h2   + N;
    float* out  = (float*)d_out;

    k_init     <<<(N * HID + 255) / 256, 256, 0, stream>>>(deg, agg1, agg2, N);
    k_degree   <<<(E + 255) / 256,       256, 0, stream>>>(col, E, deg);
    k_rsqrt    <<<(N + 255) / 256,       256, 0, stream>>>(deg, N);
    k_gemm1    <<<(N + 63) / 64,         128, 0, stream>>>(x, W1, h1, N);
    k_scatter1 <<<(E + 15) / 16,         256, 0, stream>>>(row, col, deg, h1, agg1, E);
    k_relu_bias<<<(N * HID + 255) / 256, 256, 0, stream>>>(agg1, h1, deg, b1, N);
    k_gemm2    <<<(N + 255) / 256,       256, 0, stream>>>(agg1, W2, h2, N);
    k_scatter2 <<<(E + 255) / 256,       256, 0, stream>>>(row, col, deg, h2, agg2, E);
    k_final    <<<(N + 255) / 256,       256, 0, stream>>>(agg2, h2, deg, b2, out, N);
}